// DSSA_87917980549388
// MI455X (gfx1250) — compile-verified
//
#include <hip/hip_runtime.h>
#include <math.h>

// ---------------------------------------------------------------------------
// DSSA (window attention + window-token attention) for MI455X / gfx1250.
// fp32 end-to-end; all large GEMMs on V_WMMA_F32_16X16X4_F32 (wave32 WMMA).
// B-operands staged via GLOBAL_LOAD_ASYNC_TO_LDS (ASYNCcnt) into LDS with
// bank-conflict-free strides; A-operands stream from L2-resident weights.
// Inner GEMM loops are manually software-pipelined (rotating fragment regs)
// so multiple load groups stay in flight ahead of each WMMA.
// ---------------------------------------------------------------------------

#define HEADS   16
#define DHEAD   32
#define DIM     512
#define BATCH   16
#define HH      56
#define NW      64        // windows per image (8x8)
#define NWIN    1024      // total windows
#define NTOK    50        // 1 window token + 49 pixels
#define NTOKP   64        // padded tokens for WMMA tiling
#define NPIX    49

typedef __attribute__((ext_vector_type(2))) float v2f;
typedef __attribute__((ext_vector_type(8))) float v8f;

__device__ __forceinline__ v8f wmma4(v8f c, v2f a, v2f b) {
  // D = A(16x4,f32) * B(4x16,f32) + C(16x16,f32)
  return __builtin_amdgcn_wmma_f32_16x16x4_f32(false, a, false, b, (short)0, c,
                                               false, false);
}

// Async global->LDS copies (tracked by ASYNCcnt, no VGPR round trip).
// LDS destination = low 32 bits of the generic shared pointer (LDS aperture:
// addr[31:0] is the LDS byte offset, CDNA5 ISA 10.2).
__device__ __forceinline__ void async_copy_b128(const float* smem_dst,
                                                const float* gsrc) {
  uint32_t loff = (uint32_t)(uintptr_t)smem_dst;
  asm volatile("global_load_async_to_lds_b128 %0, %1, off"
               :: "v"(loff), "v"(gsrc) : "memory");
}
__device__ __forceinline__ void async_copy_b32(const float* smem_dst,
                                               const float* gsrc) {
  uint32_t loff = (uint32_t)(uintptr_t)smem_dst;
  asm volatile("global_load_async_to_lds_b32 %0, %1, off"
               :: "v"(loff), "v"(gsrc) : "memory");
}
__device__ __forceinline__ void async_wait0() {
  asm volatile("s_wait_asynccnt 0" ::: "memory");
}

// ---------------------------------------------------------------------------
// Kernel A: gather windows + prepend window token -> xc[NWIN][DIM][NTOKP]
// ---------------------------------------------------------------------------
__global__ void __launch_bounds__(256)
k_gather_xc(const float* __restrict__ x, const float* __restrict__ wtok,
            float* __restrict__ xc) {
  int idx = blockIdx.x * 256 + threadIdx.x;   // exact: NWIN*DIM*NTOKP elements
  int n = idx & 63;
  int c = (idx >> 6) & 511;
  int w = idx >> 15;
  int bi = w >> 6, wi = w & 63;
  int xi = wi >> 3, yi = wi & 7;
  float v = 0.0f;
  if (n == 0) {
    v = wtok[c];
  } else if (n <= NPIX) {
    int t = n - 1, r = t / 7, cc = t % 7;
    v = x[((bi * DIM + c) * HH + xi * 7 + r) * HH + yi * 7 + cc];
  }
  xc[idx] = v;
}

// ---------------------------------------------------------------------------
// Kernel B: per-(window, head) fused QKV GEMM (WMMA) + attention (VALU).
// A = 96 rows of w_qkv from global (L2-resident), B = xc[win] K-chunks
// async-staged into LDS (stride 72 => conflict-free). Each wave: 3 M-tiles x
// 1 N-tile; depth-3 pipelined fragment prefetch keeps ~2 k-step load groups
// in flight ahead of each WMMA triple.
// ---------------------------------------------------------------------------
__global__ void __launch_bounds__(256)
k_qkv_attn(const float* __restrict__ w_qkv, const float* __restrict__ xc,
           float* __restrict__ attn_out) {
  __shared__ float qkv[96 * 66];   // rows: q d=0..31, k 32..63, v 64..95
  __shared__ float dots[50 * 52];
  __shared__ float stg[64 * 72];   // one K-chunk of xc (64 k x 64 tokens)
  const int win = blockIdx.x, head = blockIdx.y;
  const int tid = threadIdx.x;
  const int wave = tid >> 5, lane = tid & 31;
  const int half = lane >> 4, l = lane & 15;
  const float scale = 0.17677669529663687f;   // 32^-0.5

  const float* Bp = xc + (size_t)win * (DIM * NTOKP);

  const int nt  = wave & 3;
  const int mt0 = (wave >> 2) * 3;
  const int n0  = nt * 16;
  const float* Ap0;
  const float* Ap1;
  const float* Ap2;
  {
    int m, a;
    m = (mt0 + 0) * 16 + l; a = (m >> 5) * DIM + head * DHEAD + (m & 31);
    Ap0 = w_qkv + (size_t)a * DIM;
    m = (mt0 + 1) * 16 + l; a = (m >> 5) * DIM + head * DHEAD + (m & 31);
    Ap1 = w_qkv + (size_t)a * DIM;
    m = (mt0 + 2) * 16 + l; a = (m >> 5) * DIM + head * DHEAD + (m & 31);
    Ap2 = w_qkv + (size_t)a * DIM;
  }
  v8f acc0 = {}, acc1 = {}, acc2 = {};

  for (int kc = 0; kc < DIM; kc += 64) {
    // async-stage xc chunk: 64 k-rows x 64 tokens (16KB), b128 packets
    for (int p4 = tid; p4 < 1024; p4 += 256) {
      int c4 = (p4 & 15) << 2;
      int row = p4 >> 4;
      async_copy_b128(&stg[row * 72 + c4], Bp + (kc + row) * NTOKP + c4);
    }
    async_wait0();
    __syncthreads();

#define B_LOAD(KK, bb, x0, x1, x2) {                                  \
    int k2_ = (KK) + 2 * half;                                        \
    bb.x = stg[k2_ * 72 + n0 + l]; bb.y = stg[(k2_ + 1) * 72 + n0 + l]; \
    x0.x = Ap0[kc + k2_]; x0.y = Ap0[kc + k2_ + 1];                   \
    x1.x = Ap1[kc + k2_]; x1.y = Ap1[kc + k2_ + 1];                   \
    x2.x = Ap2[kc + k2_]; x2.y = Ap2[kc + k2_ + 1]; }

    v2f bP0, a0P0, a1P0, a2P0;
    v2f bP1, a0P1, a1P1, a2P1;
    B_LOAD(0, bP0, a0P0, a1P0, a2P0);
    B_LOAD(4, bP1, a0P1, a1P1, a2P1);
#pragma unroll
    for (int k = 0; k < 64; k += 4) {
      v2f bN, a0N, a1N, a2N;
      int kn = (k + 8) & 63;            // wraps: dummy (unused) prefetch
      B_LOAD(kn, bN, a0N, a1N, a2N);
      acc0 = wmma4(acc0, a0P0, bP0);
      acc1 = wmma4(acc1, a1P0, bP0);
      acc2 = wmma4(acc2, a2P0, bP0);
      bP0 = bP1; a0P0 = a0P1; a1P0 = a1P1; a2P0 = a2P1;
      bP1 = bN;  a0P1 = a0N;  a1P1 = a1N;  a2P1 = a2N;
    }
#undef B_LOAD
    __syncthreads();   // protect stg before next chunk's async writes
  }
#pragma unroll
  for (int j = 0; j < 8; ++j) {
    int jr = j + half * 8;
    qkv[((mt0 + 0) * 16 + jr) * 66 + n0 + l] = acc0[j];
    qkv[((mt0 + 1) * 16 + jr) * 66 + n0 + l] = acc1[j];
    qkv[((mt0 + 2) * 16 + jr) * 66 + n0 + l] = acc2[j];
  }
  __syncthreads();
  // dots[i][j] = scale * sum_d q[d][i] * k[d][j]
  for (int p = tid; p < NTOK * NTOK; p += 256) {
    int i = p / NTOK, j = p % NTOK;
    float s = 0.0f;
#pragma unroll
    for (int d = 0; d < DHEAD; ++d)
      s += qkv[d * 66 + i] * qkv[(32 + d) * 66 + j];
    dots[i * 52 + j] = s * scale;
  }
  __syncthreads();
  if (tid < NTOK) {
    float m = -1e30f;
    for (int j = 0; j < NTOK; ++j) m = fmaxf(m, dots[tid * 52 + j]);
    float s = 0.0f;
    for (int j = 0; j < NTOK; ++j) {
      float e = __expf(dots[tid * 52 + j] - m);
      dots[tid * 52 + j] = e; s += e;
    }
    float inv = 1.0f / s;
    for (int j = 0; j < NTOK; ++j) dots[tid * 52 + j] *= inv;
  }
  __syncthreads();
  float* Op = attn_out + ((size_t)win * HEADS + head) * (NTOK * DHEAD);
  for (int p = tid; p < NTOK * DHEAD; p += 256) {
    int i = p >> 5, d = p & 31;
    float s = 0.0f;
    for (int j = 0; j < NTOK; ++j)
      s += dots[i * 52 + j] * qkv[(64 + d) * 66 + j];
    Op[p] = s;
  }
}

// ---------------------------------------------------------------------------
// Kernel C: window-token LayerNorm (over d=32) + exact GELU -> gf[b][h*32+d][wi]
// ---------------------------------------------------------------------------
__global__ void __launch_bounds__(256)
k_ln_gelu(const float* __restrict__ attn_out, const float* __restrict__ ln_w,
          const float* __restrict__ ln_b, float* __restrict__ gf) {
  int g = blockIdx.x * 8 + (threadIdx.x >> 5);
  int d = threadIdx.x & 31;
  int bi = g >> 10;
  int h  = (g >> 6) & 15;
  int wi = g & 63;
  float v = attn_out[((((size_t)bi * NW + wi) * HEADS + h) * NTOK) * DHEAD + d];
  float s = v, s2 = v * v;
#pragma unroll
  for (int o = 16; o > 0; o >>= 1) {
    s  += __shfl_xor(s,  o, 32);
    s2 += __shfl_xor(s2, o, 32);
  }
  float mu  = s * (1.0f / 32.0f);
  float var = s2 * (1.0f / 32.0f) - mu * mu;
  float nrm = (v - mu) * rsqrtf(var + 1e-5f) * ln_w[d] + ln_b[d];
  float ge  = 0.5f * nrm * (1.0f + erff(nrm * 0.70710678118654752f));
  gf[((size_t)bi * DIM + h * DHEAD + d) * NW + wi] = ge;
}

// ---------------------------------------------------------------------------
// Kernel D: qk = w_qk(1024x512) @ gf(512x64) + b_qk, per batch. One WMMA
// 16x16 tile per wave (16 batches x 64 Mtiles x 4 Ntiles = 4096 waves).
// ---------------------------------------------------------------------------
__global__ void __launch_bounds__(256)
k_qk_gemm(const float* __restrict__ w_qk, const float* __restrict__ b_qk,
          const float* __restrict__ gf, float* __restrict__ qk_out) {
  int gw = blockIdx.x * 8 + (threadIdx.x >> 5);
  int lane = threadIdx.x & 31, half = lane >> 4, l = lane & 15;
  int bi = gw >> 8;
  int rem = gw & 255;
  int m0 = (rem >> 2) * 16, n0 = (rem & 3) * 16;
  const float* Ap = w_qk + (size_t)(m0 + l) * DIM;
  const float* Bp = gf + (size_t)bi * (DIM * NW);
  v8f acc = {};
  v2f a0, b0, a1, b1;
#define D_LOAD(KK, aa, bb) { int k2_ = (KK) + 2 * half;               \
    aa.x = Ap[k2_]; aa.y = Ap[k2_ + 1];                               \
    bb.x = Bp[k2_ * NW + n0 + l]; bb.y = Bp[(k2_ + 1) * NW + n0 + l]; }
  D_LOAD(0, a0, b0);
#pragma unroll 8
  for (int k = 0; k < DIM; k += 4) {
    int kn = (k + 4) & 511;
    D_LOAD(kn, a1, b1);
    acc = wmma4(acc, a0, b0);
    a0 = a1; b0 = b1;
  }
#undef D_LOAD
#pragma unroll
  for (int j = 0; j < 8; ++j) {
    int row = m0 + j + half * 8;
    qk_out[((size_t)bi * 1024 + row) * NW + n0 + l] = acc[j] + b_qk[row];
  }
}

// ---------------------------------------------------------------------------
// Kernel E: per-(b, h) window-token attention (VALU dots/softmax in LDS),
// then agg = attn(64x64) @ wfm(64 x 1568) via WMMA. wfm N-chunks are
// async-staged into LDS (stride 136 => conflict-free); each wave owns one
// N-tile and reuses its B fragment across all 4 M-tiles.
// ---------------------------------------------------------------------------
__global__ void __launch_bounds__(256)
k_wattn(const float* __restrict__ qk_out, const float* __restrict__ attn_out,
        float* __restrict__ agg) {
  // smem (dwords): attn 64x68 = 4352 | stage 64x136 = 8704 (hosts wq/wk early)
  __shared__ float smem[4352 + 64 * 136];
  float* attn = smem;
  float* stg  = smem + 4352;
  float* wq   = smem + 4352;          // 64x33 = 2112 (dead after dots)
  float* wk   = smem + 4352 + 2112;   // 64x33
  const int bi = blockIdx.x, h = blockIdx.y;
  const int tid = threadIdx.x;
  const float scale = 0.17677669529663687f;

  for (int p = tid; p < 64 * 32; p += 256) {
    int wi = p & 63, d = p >> 6;
    wq[wi * 33 + d] = qk_out[((size_t)bi * 1024 + h * 64 + d) * NW + wi];
    wk[wi * 33 + d] = qk_out[((size_t)bi * 1024 + h * 64 + 32 + d) * NW + wi];
  }
  __syncthreads();
  for (int p = tid; p < 64 * 64; p += 256) {
    int i = p >> 6, j = p & 63;
    float s = 0.0f;
#pragma unroll
    for (int d = 0; d < 32; ++d) s += wq[i * 33 + d] * wk[j * 33 + d];
    attn[i * 68 + j] = s * scale;
  }
  __syncthreads();
  if (tid < 64) {
    float m = -1e30f;
    for (int j = 0; j < 64; ++j) m = fmaxf(m, attn[tid * 68 + j]);
    float s = 0.0f;
    for (int j = 0; j < 64; ++j) {
      float e = __expf(attn[tid * 68 + j] - m);
      attn[tid * 68 + j] = e; s += e;
    }
    float inv = 1.0f / s;
    for (int j = 0; j < 64; ++j) attn[tid * 68 + j] *= inv;
  }
  __syncthreads();

  const int wave = tid >> 5, lane = tid & 31, half = lane >> 4, l = lane & 15;
  const float* Bp = attn_out + ((size_t)bi * NW * HEADS + h) * (NTOK * DHEAD)
                    + DHEAD;                    // skip token 0 -> wfm
  const int ldb = HEADS * NTOK * DHEAD;         // stride between windows j
  float* Dp = agg + ((size_t)bi * HEADS + h) * (NW * NPIX * DHEAD);

  for (int nc = 0; nc < NW * NPIX / 2; nc += 128) {   // 1568 cols, 13 chunks
    int ncols = (NW * NPIX / 2) - nc; if (ncols > 128) ncols = 128;
    int cpr = ncols >> 2;                       // b128 packets per row
    // async-stage wfm chunk: 64 j-rows x ncols
    for (int p4 = tid; p4 < 64 * cpr; p4 += 256) {
      int c4 = (p4 % cpr) << 2;
      int j  = p4 / cpr;
      async_copy_b128(&stg[j * 136 + c4], Bp + (size_t)j * ldb + nc + c4);
    }
    async_wait0();
    __syncthreads();
    int ntiles = ncols >> 4;                    // 8 or 2 N-tiles
    if (wave < ntiles) {
      int n0 = wave * 16;
      v8f a0 = {}, a1 = {}, a2 = {}, a3 = {};
      for (int k = 0; k < 64; k += 4) {
        int k2 = k + 2 * half;
        v2f b = { stg[k2 * 136 + n0 + l], stg[(k2 + 1) * 136 + n0 + l] };
        v2f a;
        a = *(const v2f*)&attn[(0  + l) * 68 + k2]; a0 = wmma4(a0, a, b);
        a = *(const v2f*)&attn[(16 + l) * 68 + k2]; a1 = wmma4(a1, a, b);
        a = *(const v2f*)&attn[(32 + l) * 68 + k2]; a2 = wmma4(a2, a, b);
        a = *(const v2f*)&attn[(48 + l) * 68 + k2]; a3 = wmma4(a3, a, b);
      }
#pragma unroll
      for (int j = 0; j < 8; ++j) {
        int jr = j + half * 8;
        Dp[(size_t)(0  + jr) * (NPIX * DHEAD) + nc + n0 + l] = a0[j];
        Dp[(size_t)(16 + jr) * (NPIX * DHEAD) + nc + n0 + l] = a1[j];
        Dp[(size_t)(32 + jr) * (NPIX * DHEAD) + nc + n0 + l] = a2[j];
        Dp[(size_t)(48 + jr) * (NPIX * DHEAD) + nc + n0 + l] = a3[j];
      }
    }
    __syncthreads();   // all reads of stg done before next chunk's stage
  }
}

// ---------------------------------------------------------------------------
// Kernel F: out = w_out(512x512) @ fmap + b_out, per (batch, window, m-half).
// B tile (64 channels x 49 tokens, zero-padded to 64) async-scatter-staged
// from agg into LDS (per-lane LDS addresses; stride 72 => conflict-free).
// k-outer compute: per k-step 2 A + 4 B fragments feed 8 WMMAs (depth-2
// pipelined). Scatter-store to NCHW output.
// ---------------------------------------------------------------------------
__global__ void __launch_bounds__(256)
k_out_gemm(const float* __restrict__ w_out, const float* __restrict__ b_out,
           const float* __restrict__ agg, float* __restrict__ out) {
  __shared__ float bl[64 * 72];
  const int bi = blockIdx.x;           // batch
  const int wi = blockIdx.y;           // window
  const int mh = blockIdx.z;           // output-row half (o base = mh*256)
  const int xi = wi >> 3, yi = wi & 7;
  const int tid = threadIdx.x;
  const int wave = tid >> 5, lane = tid & 31, half = lane >> 4, l = lane & 15;

  // zero once; chunks only overwrite n < 49, so n in [49,64) stays zero
  for (int p = tid; p < 64 * 72; p += 256) bl[p] = 0.0f;
  __syncthreads();

  v8f acc[8] = {};                     // [mi*4 + ni]
  const float* Ag = agg + (size_t)bi * (HEADS * NW * NPIX * DHEAD);
  const int m0A = mh * 256 + (wave * 2 + 0) * 16;
  const int m0B = mh * 256 + (wave * 2 + 1) * 16;

  for (int kc = 0; kc < DIM; kc += 64) {
    // async transpose-gather: bl[c'][n] = agg[(h)(wi)][n][d], c' = c - kc
    for (int p = tid; p < 64 * 64; p += 256) {
      int cp = p & 63, n = p >> 6;
      int c = kc + cp;
      if (n < NPIX)
        async_copy_b32(&bl[cp * 72 + n],
                       Ag + (size_t)((c >> 5) * NW + wi) * (NPIX * DHEAD)
                          + n * DHEAD + (c & 31));
    }
    async_wait0();
    __syncthreads();
    const float* ApA = w_out + (size_t)(m0A + l) * DIM + kc;
    const float* ApB = w_out + (size_t)(m0B + l) * DIM + kc;

#define F_LOAD(KK, aa, ab, b0, b1, b2, b3) {                          \
    int k2_ = (KK) + 2 * half;                                        \
    aa.x = ApA[k2_]; aa.y = ApA[k2_ + 1];                             \
    ab.x = ApB[k2_]; ab.y = ApB[k2_ + 1];                             \
    b0.x = bl[k2_ * 72 +  0 + l]; b0.y = bl[(k2_ + 1) * 72 +  0 + l]; \
    b1.x = bl[k2_ * 72 + 16 + l]; b1.y = bl[(k2_ + 1) * 72 + 16 + l]; \
    b2.x = bl[k2_ * 72 + 32 + l]; b2.y = bl[(k2_ + 1) * 72 + 32 + l]; \
    b3.x = bl[k2_ * 72 + 48 + l]; b3.y = bl[(k2_ + 1) * 72 + 48 + l]; }

    v2f aa0, ab0, b00, b10, b20, b30;
    F_LOAD(0, aa0, ab0, b00, b10, b20, b30);
#pragma unroll
    for (int k = 0; k < 64; k += 4) {
      v2f aa1, ab1, b01, b11, b21, b31;
      int kn = (k + 4) & 63;           // wraps: dummy (unused) prefetch
      F_LOAD(kn, aa1, ab1, b01, b11, b21, b31);
      acc[0] = wmma4(acc[0], aa0, b00);
      acc[1] = wmma4(acc[1], aa0, b10);
      acc[2] = wmma4(acc[2], aa0, b20);
      acc[3] = wmma4(acc[3], aa0, b30);
      acc[4] = wmma4(acc[4], ab0, b00);
      acc[5] = wmma4(acc[5], ab0, b10);
      acc[6] = wmma4(acc[6], ab0, b20);
      acc[7] = wmma4(acc[7], ab0, b30);
      aa0 = aa1; ab0 = ab1; b00 = b01; b10 = b11; b20 = b21; b30 = b31;
    }
#undef F_LOAD
    __syncthreads();   // protect bl before next chunk's async writes
  }
#pragma unroll
  for (int mi = 0; mi < 2; ++mi) {
    int m0 = mh * 256 + (wave * 2 + mi) * 16;
#pragma unroll
    for (int ni = 0; ni < 4; ++ni) {
      int n = ni * 16 + l;
      if (n < NPIX) {
        int r = n / 7, cc = n % 7;
        int hh = xi * 7 + r, ww = yi * 7 + cc;
#pragma unroll
        for (int j = 0; j < 8; ++j) {
          int o = m0 + j + half * 8;
          out[((size_t)(bi * DIM + o) * HH + hh) * HH + ww] =
              acc[mi * 4 + ni][j] + b_out[o];
        }
      }
    }
  }
}

// ---------------------------------------------------------------------------
// Host launcher. Workspace layout (floats); agg aliases dead xc region.
//   xc       : [0,            33,554,432)
//   attn_out : [33,554,432,   59,768,832)
//   gf       : [59,768,832,   60,293,120)
//   qk_out   : [60,293,120,   61,341,696)
//   agg      : [0,            25,690,112)   (reuses xc after kernel B)
// Total: ~234 MB of d_ws.
// ---------------------------------------------------------------------------
extern "C" void kernel_launch(void* const* d_in, const int* in_sizes, int n_in,
                              void* d_out, int out_size, void* d_ws,
                              size_t ws_size, hipStream_t stream) {
  const float* x      = (const float*)d_in[0];
  const float* w_qkv  = (const float*)d_in[1];
  const float* wtok   = (const float*)d_in[2];
  const float* ln_w   = (const float*)d_in[3];
  const float* ln_b   = (const float*)d_in[4];
  const float* w_qk   = (const float*)d_in[5];
  const float* b_qk   = (const float*)d_in[6];
  const float* w_out  = (const float*)d_in[7];
  const float* b_out  = (const float*)d_in[8];
  float* out = (float*)d_out;
  float* ws  = (float*)d_ws;

  float* xc       = ws;
  float* attn_out = ws + 33554432u;
  float* gf       = ws + 59768832u;
  float* qk_out   = ws + 60293120u;
  float* agg      = ws;              // alias xc (dead after k_qkv_attn)

  k_gather_xc<<<131072, 256, 0, stream>>>(x, wtok, xc);
  k_qkv_attn<<<dim3(NWIN, HEADS), 256, 0, stream>>>(w_qkv, xc, attn_out);
  k_ln_gelu<<<2048, 256, 0, stream>>>(attn_out, ln_w, ln_b, gf);
  k_qk_gemm<<<512, 256, 0, stream>>>(w_qk, b_qk, gf, qk_out);
  k_wattn<<<dim3(BATCH, HEADS), 256, 0, stream>>>(qk_out, attn_out, agg);
  k_out_gemm<<<dim3(BATCH, NW, 2), 256, 0, stream>>>(w_out, b_out, agg, out);
}